// And_Convolution_56564719288841
// MI455X (gfx1250) — compile-verified
//
#include <hip/hip_runtime.h>

#define KSIZE   16
#define STRIDE  4
#define TILE_OUT 512
#define BLOCK   256
// input floats a full tile needs: TILE_OUT*STRIDE + (KSIZE-STRIDE) = 2060
#define TILE_FLOATS (TILE_OUT * STRIDE + (KSIZE - STRIDE))

#ifdef __has_builtin
#if __has_builtin(__builtin_amdgcn_global_load_async_to_lds_b128)
#define HAVE_ASYNC_LDS 1
#endif
#if __has_builtin(__builtin_amdgcn_s_wait_asynccnt)
#define HAVE_WAIT_ASYNC 1
#endif
#endif

#define AS_GLOBAL __attribute__((address_space(1)))
#define AS_LDS    __attribute__((address_space(3)))

typedef int v4i __attribute__((vector_size(4 * sizeof(int))));

__global__ __launch_bounds__(BLOCK) void and_conv_min_kernel(
    const float* __restrict__ inp,   // [N, L]
    const float* __restrict__ wgt,   // [1, KSIZE]
    float* __restrict__ out,         // [N, nOut]
    int L, int nOutPerRow)
{
    __shared__ __align__(16) float tile[TILE_FLOATS + 4];

    const int row = blockIdx.y;
    const int s0  = blockIdx.x * TILE_OUT;                 // first output of this tile
    const int nOut = min(TILE_OUT, nOutPerRow - s0);       // outputs in this tile
    const int tileFloats = nOut * STRIDE + (KSIZE - STRIDE); // always multiple of 4
    const int nVec = tileFloats >> 2;                      // float4 count

    const float* gbase = inp + (size_t)row * (size_t)L + (size_t)s0 * STRIDE;

    // ---- stage global -> LDS (CDNA5 async path when available) ----
#if defined(HAVE_ASYNC_LDS)
    for (int v = threadIdx.x; v < nVec; v += BLOCK) {
        __builtin_amdgcn_global_load_async_to_lds_b128(
            (AS_GLOBAL v4i*)(gbase + 4 * v),
            (AS_LDS    v4i*)(&tile[4 * v]),
            /*offset=*/0, /*cpol=*/0);
    }
#if defined(HAVE_WAIT_ASYNC)
    __builtin_amdgcn_s_wait_asynccnt(0);
#else
    asm volatile("s_wait_asynccnt 0" ::: "memory");
#endif
#else
    for (int v = threadIdx.x; v < nVec; v += BLOCK) {
        float4 x = *(const float4*)(gbase + 4 * v);
        *(float4*)(&tile[4 * v]) = x;
    }
#endif
    __syncthreads();

    // ---- weights into registers (uniform -> scalar loads) ----
    float w[KSIZE];
#pragma unroll
    for (int k = 0; k < KSIZE; ++k) w[k] = wgt[k];

    float* outRow = out + (size_t)row * (size_t)nOutPerRow + s0;

    // ---- each thread computes its outputs from LDS ----
    for (int j = threadIdx.x; j < nOut; j += BLOCK) {
        const float4* p = (const float4*)(&tile[j * STRIDE]); // 16B aligned
        float4 x0 = p[0];
        float4 x1 = p[1];
        float4 x2 = p[2];
        float4 x3 = p[3];

        float m0 = fminf(fminf(x0.x * w[0],  x0.y * w[1]),
                         fminf(x0.z * w[2],  x0.w * w[3]));
        float m1 = fminf(fminf(x1.x * w[4],  x1.y * w[5]),
                         fminf(x1.z * w[6],  x1.w * w[7]));
        float m2 = fminf(fminf(x2.x * w[8],  x2.y * w[9]),
                         fminf(x2.z * w[10], x2.w * w[11]));
        float m3 = fminf(fminf(x3.x * w[12], x3.y * w[13]),
                         fminf(x3.z * w[14], x3.w * w[15]));

        outRow[j] = fminf(fminf(m0, m1), fminf(m2, m3));
    }
}

extern "C" void kernel_launch(void* const* d_in, const int* in_sizes, int n_in,
                              void* d_out, int out_size, void* d_ws, size_t ws_size,
                              hipStream_t stream) {
    (void)n_in; (void)d_ws; (void)ws_size;

    const float* inp = (const float*)d_in[0];   // [N, L] fp32
    const float* wgt = (const float*)d_in[1];   // [1, 16] fp32

    const int L = 8192;
    const int N = in_sizes[0] / L;
    const int nOutPerRow = (L - KSIZE) / STRIDE + 1;  // 2045
    (void)out_size;

    dim3 grid((nOutPerRow + TILE_OUT - 1) / TILE_OUT, N, 1);
    dim3 block(BLOCK, 1, 1);
    and_conv_min_kernel<<<grid, block, 0, stream>>>(inp, wgt, (float*)d_out,
                                                    L, nOutPerRow);
}